// GCN_Encoder_27917287424811
// MI455X (gfx1250) — compile-verified
//
#include <hip/hip_runtime.h>
#include <math.h>

typedef float v2f __attribute__((ext_vector_type(2)));
typedef float v8f __attribute__((ext_vector_type(8)));

#define DIM 128
#define BN_EPS 1e-5f

// Shared-memory layout for the GEMM (single buffer, W region FIRST so the hot
// inner-loop ds loads use small immediate offsets):
//   W stored as K-pairs: pair p = k>>1 holds (W[2p][n], W[2p+1][n]) as float2
//   at smem2[p * SP + n].  SP = 144 float2  ->  lane-half delta 288 dwords
//   == 32 (mod 64): lanes 0-15 use banks 0-31, lanes 16-31 banks 32-63.
#define SP 144                        // pair-row stride (float2 units)
#define SW_F2 (64 * SP)               // W region size in float2 (9216)
#define SA_OFF (SW_F2 * 2)            // float offset of A region (18432 floats)
#define SA_STRIDE 132                 // A row stride in floats (conflict-free)
#define SMEM_FLOATS (SA_OFF + DIM * SA_STRIDE)   // 35328 floats = 138 KB

// ---------------------------------------------------------------------------
// Utility kernels
// ---------------------------------------------------------------------------
__global__ void zero_kernel(float* __restrict__ p, size_t n) {
    size_t i = (size_t)blockIdx.x * blockDim.x + threadIdx.x;
    size_t stride = (size_t)gridDim.x * blockDim.x;
    for (; i < n; i += stride) p[i] = 0.0f;
}

// Count degrees for all relations with f32 atomics. edges layout [rel][2][nE].
__global__ void degree_kernel(const int* __restrict__ edges,
                              float* __restrict__ degOut, float* __restrict__ degIn,
                              int nE, int nRel, int nNodes) {
    size_t total = (size_t)nRel * nE;
    size_t i = (size_t)blockIdx.x * blockDim.x + threadIdx.x;
    size_t stride = (size_t)gridDim.x * blockDim.x;
    for (; i < total; i += stride) {
        int r = (int)(i / nE);
        int e = (int)(i % nE);
        int s = edges[((size_t)r * 2 + 0) * nE + e];
        int d = edges[((size_t)r * 2 + 1) * nE + e];
        atomicAdd(&degOut[(size_t)r * nNodes + s], 1.0f);
        atomicAdd(&degIn[(size_t)r * nNodes + d], 1.0f);
    }
}

// deg -> rsqrt(max(deg, 1))
__global__ void rsqrt_clip_kernel(float* __restrict__ p, size_t n) {
    size_t i = (size_t)blockIdx.x * blockDim.x + threadIdx.x;
    size_t stride = (size_t)gridDim.x * blockDim.x;
    for (; i < n; i += stride) p[i] = rsqrtf(fmaxf(p[i], 1.0f));
}

// agg[i, c] = sum over 3 relations of conv_b[l, r, c]
__global__ void init_agg_kernel(float* __restrict__ agg, const float* __restrict__ b3,
                                size_t total) {
    size_t i = (size_t)blockIdx.x * blockDim.x + threadIdx.x;
    size_t stride = (size_t)gridDim.x * blockDim.x;
    for (; i < total; i += stride) {
        int c = (int)(i & (DIM - 1));
        agg[i] = b3[c] + b3[DIM + c] + b3[2 * DIM + c];
    }
}

// One wave (32 lanes) per edge: m[dst] += h[src] * dout_rs[src]  (128 floats)
__global__ void scatter_kernel(const float* __restrict__ h,
                               const int* __restrict__ src, const int* __restrict__ dst,
                               const float* __restrict__ dout_rs,
                               float* __restrict__ m, int nE) {
    int gw = (int)(((size_t)blockIdx.x * blockDim.x + threadIdx.x) >> 5);
    int lane = threadIdx.x & 31;
    if (gw >= nE) return;
    int s = src[gw];
    int d = dst[gw];
    float sc = dout_rs[s];
    float4 hv = *(const float4*)(h + (size_t)s * DIM + lane * 4);
    float* mp = m + (size_t)d * DIM + lane * 4;
    atomicAdd(mp + 0, hv.x * sc);
    atomicAdd(mp + 1, hv.y * sc);
    atomicAdd(mp + 2, hv.z * sc);
    atomicAdd(mp + 3, hv.w * sc);
}

// ---------------------------------------------------------------------------
// fp32 WMMA GEMM:  Cout[N x 128] = op( (rowScale .* A)[N x 128] @ W[128 x 128] )
//   SCALE : multiply each A row by rowScale[row]
//   ACCUM : accumulator initialized from Cin (agg accumulation)  else from bias
//   RELU  : relu epilogue
// Block = 256 threads = 8 waves; each wave owns a 16-row x 128-col strip.
// B fragment = single conflict-free ds_load_b64 with immediate offset.
// ---------------------------------------------------------------------------
template <bool SCALE, bool ACCUM, bool RELU>
__launch_bounds__(256)
__global__ void gemm128_kernel(const float* __restrict__ A,
                               const float* __restrict__ rowScale,
                               const float* __restrict__ W,
                               const float* __restrict__ bias,
                               const float* __restrict__ Cin,
                               float* __restrict__ Cout,
                               int nRows) {
    __shared__ float smem[SMEM_FLOATS];

    const int tid = threadIdx.x;
    const int blockRow0 = blockIdx.x * 128;

    // Stage W into LDS in K-pair layout (coalesced float4 global reads).
    {
        int n4 = (tid & 31) * 4;
        int krow = tid >> 5;             // 0..7
#pragma unroll
        for (int i = 0; i < 16; ++i) {
            int k = krow + i * 8;        // 0..127
            float4 v = *(const float4*)(W + (size_t)k * DIM + n4);
            float* dstp = smem + ((k >> 1) * SP + n4) * 2 + (k & 1);
            dstp[0] = v.x;
            dstp[2] = v.y;
            dstp[4] = v.z;
            dstp[6] = v.w;
        }
    }
    // Stage A block (up to 128 rows), applying the per-row scale.
    {
        int col = (tid & 31) * 4;
        int rowBase = tid >> 5;          // 0..7
#pragma unroll
        for (int i = 0; i < 16; ++i) {
            int lrow = rowBase + i * 8;  // 0..127
            int grow = blockRow0 + lrow;
            float4 v = make_float4(0.f, 0.f, 0.f, 0.f);
            float sc = 1.0f;
            if (grow < nRows) {
                v = *(const float4*)(A + (size_t)grow * DIM + col);
                if (SCALE) sc = rowScale[grow];
            }
            float* dstp = smem + SA_OFF + lrow * SA_STRIDE + col;
            dstp[0] = v.x * sc;
            dstp[1] = v.y * sc;
            dstp[2] = v.z * sc;
            dstp[3] = v.w * sc;
        }
    }
    __syncthreads();

    const int wave = tid >> 5;
    const int lane = tid & 31;
    const int laneLo = lane & 15;
    const int hi = lane >> 4;            // lane half selects K pair

    const int tileRow0 = blockRow0 + wave * 16;
    if (tileRow0 >= nRows) return;       // nRows is a multiple of 16 here

    // Load the 32 A fragments (16x4 each) for this wave's rows from LDS.
    // ISA layout: lanes 0-15 hold K={k0,k0+1}, lanes 16-31 hold K={k0+2,k0+3}.
    const int lrow = wave * 16 + laneLo;
    v2f a[32];
    {
        const float* ap = smem + SA_OFF + lrow * SA_STRIDE + 2 * hi;
#pragma unroll
        for (int kk = 0; kk < 32; ++kk)
            a[kk] = *(const v2f*)(ap + 4 * kk);
    }

    const v2f* smem2 = (const v2f*)smem;

#pragma unroll
    for (int nt = 0; nt < 8; ++nt) {
        const int n = nt * 16 + laneLo;
        // Pair index for k-step kk is (2*kk + hi); two bases keep every ds
        // immediate under 64 KB.
        const v2f* wp0 = smem2 + hi * SP + n;     // kk in [0,16)
        const v2f* wp1 = wp0 + 32 * SP;           // kk in [16,32)

        v8f c;
        if (ACCUM) {
#pragma unroll
            for (int v = 0; v < 8; ++v)
                c[v] = Cin[(size_t)(tileRow0 + v + 8 * hi) * DIM + n];
        } else {
            float bv = bias[n];
#pragma unroll
            for (int v = 0; v < 8; ++v) c[v] = bv;
        }
#pragma unroll
        for (int kk = 0; kk < 32; ++kk) {
            v2f b = (kk < 16) ? wp0[2 * kk * SP] : wp1[2 * (kk - 16) * SP];
            // (neg_a, A, neg_b, B, c_mod, C, reuse_a, reuse_b)
            c = __builtin_amdgcn_wmma_f32_16x16x4_f32(false, a[kk], false, b,
                                                      (short)0, c, false, false);
        }
#pragma unroll
        for (int v = 0; v < 8; ++v) {
            float o = c[v];
            if (RELU) o = fmaxf(o, 0.0f);
            Cout[(size_t)(tileRow0 + v + 8 * hi) * DIM + n] = o;
        }
    }
}

// ---------------------------------------------------------------------------
// BatchNorm (training-mode batch stats) over nodes, per feature column.
// sums[0..127] = sum, sums[128..255] = sum of squares.
// ---------------------------------------------------------------------------
__global__ void bn_stats_kernel(const float* __restrict__ h, float* __restrict__ sums,
                                int nNodes) {
    int c = threadIdx.x & (DIM - 1);
    int rowsPerBlock = blockDim.x >> 7;
    int row = blockIdx.x * rowsPerBlock + (threadIdx.x >> 7);
    int strideRows = gridDim.x * rowsPerBlock;
    float s = 0.0f, s2 = 0.0f;
    for (int i = row; i < nNodes; i += strideRows) {
        float v = h[(size_t)i * DIM + c];
        s += v;
        s2 += v * v;
    }
    atomicAdd(&sums[c], s);
    atomicAdd(&sums[DIM + c], s2);
}

__global__ void bn_apply_kernel(float* __restrict__ h, const float* __restrict__ sums,
                                const float* __restrict__ gamma,
                                const float* __restrict__ beta, int nNodes) {
    size_t total = (size_t)nNodes * DIM;
    float invN = 1.0f / (float)nNodes;
    size_t i = (size_t)blockIdx.x * blockDim.x + threadIdx.x;
    size_t stride = (size_t)gridDim.x * blockDim.x;
    for (; i < total; i += stride) {
        int c = (int)(i & (DIM - 1));
        float mean = sums[c] * invN;
        float var = sums[DIM + c] * invN - mean * mean;
        float inv = rsqrtf(var + BN_EPS);
        h[i] = (h[i] - mean) * inv * gamma[c] + beta[c];
    }
}

// ---------------------------------------------------------------------------
// Host orchestration
// ---------------------------------------------------------------------------
extern "C" void kernel_launch(void* const* d_in, const int* in_sizes, int n_in,
                              void* d_out, int out_size, void* d_ws, size_t ws_size,
                              hipStream_t stream) {
    const float* x        = (const float*)d_in[0];  // [N,128]
    const int*   edges    = (const int*)  d_in[1];  // [3,2,E]
    const float* conv_W   = (const float*)d_in[2];  // [3,3,128,128]
    const float* conv_b   = (const float*)d_in[3];  // [3,3,128]
    const float* fc_W     = (const float*)d_in[4];  // [3,128,128]
    const float* fc_b     = (const float*)d_in[5];  // [3,128]
    const float* bn_gamma = (const float*)d_in[6];  // [3,128]
    const float* bn_beta  = (const float*)d_in[7];  // [3,128]
    float* out = (float*)d_out;

    const int N = in_sizes[0] / DIM;     // 100000
    const int E = in_sizes[1] / (3 * 2); // 600000
    const int R = 3;

    // Workspace carve-up.
    float* hA     = (float*)d_ws;                        // N*128
    float* mbuf   = hA + (size_t)N * DIM;                // N*128
    float* agg    = mbuf + (size_t)N * DIM;              // N*128
    float* degOut = agg + (size_t)N * DIM;               // 3*N (then degIn 3*N)
    float* degIn  = degOut + (size_t)3 * N;
    float* sums   = degIn + (size_t)3 * N;               // 256

    const int gemmBlocks = ((N + 15) / 16 + 7) / 8;      // 782
    const int scatterBlocks = (int)(((size_t)E * 32 + 255) / 256);

    // Degrees are layer-invariant: count once per call, convert to rsqrt.
    zero_kernel<<<1024, 256, 0, stream>>>(degOut, (size_t)6 * N);
    degree_kernel<<<2048, 256, 0, stream>>>(edges, degOut, degIn, E, R, N);
    rsqrt_clip_kernel<<<1024, 256, 0, stream>>>(degOut, (size_t)6 * N);

    const float* hin = x;
    for (int l = 0; l < 3; ++l) {
        // L0 -> hA, L1 -> d_out, L2 -> d_out (in place after agg is built)
        float* hout = (l == 0) ? hA : out;

        init_agg_kernel<<<2048, 256, 0, stream>>>(agg, conv_b + (size_t)l * R * DIM,
                                                  (size_t)N * DIM);
        for (int r = 0; r < R; ++r) {
            zero_kernel<<<2048, 256, 0, stream>>>(mbuf, (size_t)N * DIM);
            scatter_kernel<<<scatterBlocks, 256, 0, stream>>>(
                hin,
                edges + ((size_t)r * 2 + 0) * E,
                edges + ((size_t)r * 2 + 1) * E,
                degOut + (size_t)r * N, mbuf, E);
            // agg += (din .* m) @ conv_W[l,r]
            gemm128_kernel<true, true, false><<<gemmBlocks, 256, 0, stream>>>(
                mbuf, degIn + (size_t)r * N,
                conv_W + ((size_t)l * R + r) * DIM * DIM,
                nullptr, agg, agg, N);
        }
        // h = relu(agg @ fc_W[l] + fc_b[l])
        gemm128_kernel<false, false, true><<<gemmBlocks, 256, 0, stream>>>(
            agg, nullptr, fc_W + (size_t)l * DIM * DIM,
            fc_b + (size_t)l * DIM, nullptr, hout, N);

        // BatchNorm with batch statistics.
        zero_kernel<<<1, 256, 0, stream>>>(sums, (size_t)256);
        bn_stats_kernel<<<512, 256, 0, stream>>>(hout, sums, N);
        bn_apply_kernel<<<2048, 256, 0, stream>>>(hout, sums,
                                                  bn_gamma + (size_t)l * DIM,
                                                  bn_beta + (size_t)l * DIM, N);
        hin = hout;
    }
}